// EdgeNetwork_31808527794434
// MI455X (gfx1250) — compile-verified
//
#include <hip/hip_runtime.h>

typedef __attribute__((ext_vector_type(2))) float v2f;
typedef __attribute__((ext_vector_type(4))) float v4f;
typedef __attribute__((ext_vector_type(8))) float v8f;

#define N_NODES 50000
#define N_EDGES 800000
#define DIM     16
#define NCHUNK  68              // K = 272 = 256 (kernel) + 16 (bias rows), 4 per WMMA
#define NGROUP  (N_EDGES / 16)  // 16 edges per wave-group

__global__ void edgenet_zero(float* __restrict__ out, int n) {
    int i = blockIdx.x * blockDim.x + threadIdx.x;
    if (i < n) out[i] = 0.0f;
}

__global__ __launch_bounds__(256)
void edgenet_wmma(const float* __restrict__ nodef,
                  const float* __restrict__ edgef,
                  const long long* __restrict__ pair,
                  const float* __restrict__ kern,
                  const float* __restrict__ bias,
                  float* __restrict__ out) {
    // B-matrix (272 x 16): rows 0..255 = kernel^T (B[k*16+j][i] = kernel[k, i*16+j]),
    // rows 256..271 = bias   (B[256+j][i]   = bias[i*16+j]).
    // Stored in LDS pre-swizzled into the 4x16 f32 WMMA B-fragment layout:
    // chunk c, lane l: {row 4c+kb+0, row 4c+kb+1} at column l&15, kb=(l<16)?0:2.
    __shared__ v2f ldsB[NCHUNK * 32];

    const int tid = threadIdx.x;
    for (int s = tid; s < NCHUNK * 32; s += blockDim.x) {
        const int c  = s >> 5;
        const int l  = s & 31;
        const int i  = l & 15;
        const int kb = (l < 16) ? 0 : 2;
        v2f bv;
        #pragma unroll
        for (int t = 0; t < 2; ++t) {
            const int kj = 4 * c + kb + t;
            bv[t] = (kj < 256) ? kern[(kj >> 4) * 256 + i * 16 + (kj & 15)]
                               : bias[i * 16 + (kj - 256)];
        }
        ldsB[s] = bv;
    }
    __syncthreads();

    const int lane = tid & 31;
    const int wave = tid >> 5;
    const int gw   = blockIdx.x * (blockDim.x >> 5) + wave;
    const int nw   = gridDim.x * (blockDim.x >> 5);

    for (int g = gw; g < NGROUP; g += nw) {
        // lanes l and l+16 both service edge g*16 + (l&15)
        const long long e   = (long long)g * 16 + (lane & 15);
        const long long src = pair[2 * e + 0];
        const long long dst = pair[2 * e + 1];

        float ef[16], nb[16];
        const v4f* efp = (const v4f*)(edgef + e * 16);
        const v4f* nbp = (const v4f*)(nodef + dst * 16);
        #pragma unroll
        for (int q = 0; q < 4; ++q) {
            v4f a = efp[q];
            v4f b = nbp[q];
            #pragma unroll
            for (int t = 0; t < 4; ++t) { ef[4 * q + t] = a[t]; nb[4 * q + t] = b[t]; }
        }

        // Pre-shift nb by this lane's K offset so all array indices below are
        // compile-time constants (keeps ef/nbs in VGPRs). Only m=0..13 are used.
        float nbs[14];
        #pragma unroll
        for (int m = 0; m < 14; ++m)
            nbs[m] = (lane < 16) ? nb[m] : nb[m + 2];

        v8f acc = {};
        #pragma unroll
        for (int c = 0; c < NCHUNK; ++c) {
            v2f a;
            if (c < 64) {
                // outer-product entries: kk = 4c + kbase + t
                // ef index (kk>>4) == c>>2 for both halves; nb index absorbed in nbs
                const float efv = ef[c >> 2];
                const int   j   = (4 * c) & 15;
                a[0] = efv * nbs[j];
                a[1] = efv * nbs[j + 1];
            } else {
                // bias rows: A entries are the raw neighbor values
                const int j = 4 * (c - 64);
                a[0] = nbs[j];
                a[1] = nbs[j + 1];
            }
            const v2f b = ldsB[c * 32 + lane];
            // D(16edges x 16dims) += A(16x4 f32) x B(4x16 f32)
            acc = __builtin_amdgcn_wmma_f32_16x16x4_f32(
                false, a, false, b, (short)0, acc, false, false);
        }

        // C/D layout: VGPR v, lanes 0-15 -> edge v, lanes 16-31 -> edge v+8; col = lane&15
        const int srcI = (int)src;
        #pragma unroll
        for (int v = 0; v < 8; ++v) {
            const int sel = v + ((lane & 16) >> 1);   // v or v+8
            const int so  = __shfl(srcI, sel, 32);
            atomicAdd(out + so * 16 + (lane & 15), acc[v]);
        }
    }
}

extern "C" void kernel_launch(void* const* d_in, const int* in_sizes, int n_in,
                              void* d_out, int out_size, void* d_ws, size_t ws_size,
                              hipStream_t stream) {
    const float*     nodef = (const float*)d_in[0];     // (50000, 16) f32
    const float*     edgef = (const float*)d_in[1];     // (800000, 16) f32
    const long long* pair  = (const long long*)d_in[2]; // (800000, 2) int64
    const float*     kern  = (const float*)d_in[3];     // (16, 256) f32
    const float*     bias  = (const float*)d_in[4];     // (256,) f32
    float*           out   = (float*)d_out;             // (50000, 16) f32

    edgenet_zero<<<(N_NODES * DIM + 255) / 256, 256, 0, stream>>>(out, N_NODES * DIM);
    edgenet_wmma<<<1024, 256, 0, stream>>>(nodef, edgef, pair, kern, bias, out);
}